// RCNNCell_28570122453287
// MI455X (gfx1250) — compile-verified
//
#include <hip/hip_runtime.h>

// MI455X / gfx1250, wave32. Stencil-as-banded-matmul using V_WMMA_F32_16X16X4_F32.
typedef float v2f __attribute__((ext_vector_type(2)));
typedef float v8f __attribute__((ext_vector_type(8)));

namespace {
constexpr int   GN      = 2048;
constexpr int   GMSK    = GN - 1;         // power-of-two wrap
constexpr float DT_C    = 0.00025f;
constexpr float INV_DX  = 100.0f;         // 1/DX
constexpr float INV_DX2 = 10000.0f;       // 1/DX^2
}

// 5-tap band value by offset d in [0,4] (taps at -2..+2), else 0.
__device__ __forceinline__ float band5(int d, float w0, float w1, float w2,
                                       float w3, float w4) {
  float r = 0.0f;
  r = (d == 0) ? w0 : r;
  r = (d == 1) ? w1 : r;
  r = (d == 2) ? w2 : r;
  r = (d == 3) ? w3 : r;
  r = (d == 4) ? w4 : r;
  return r;
}

// Banded coefficient operand for K-chunk j (global K = 4j..4j+3).
// Identical register recipe for the A-side (rows m = lane&15) and the B-side
// (cols n = lane&15): val = w[k - mn] if 0 <= k-mn <= 4 else 0, where
// k = 4j + 2*(lane>=16) + vgpr  (fp32 16x4 A / 4x16 B operand = 2 VGPRs).
template <bool LAP>
__device__ __forceinline__ v2f coef_chunk(int j, int lane) {
  const int mn = lane & 15;
  const int kb = 4 * j + ((lane >> 4) << 1);
  v2f c;
#pragma unroll
  for (int i = 0; i < 2; ++i) {
    const int d = kb + i - mn;
    if (LAP)  // 1D 4th-order laplacian arm: [-1/12, 4/3, -5/2, 4/3, -1/12]/dx^2
      c[i] = band5(d, (-1.0f / 12.0f) * INV_DX2, (4.0f / 3.0f) * INV_DX2,
                   -2.5f * INV_DX2, (4.0f / 3.0f) * INV_DX2,
                   (-1.0f / 12.0f) * INV_DX2);
    else      // 4th-order central derivative: [1/12, -8/12, 0, 8/12, -1/12]/dx
      c[i] = band5(d, (1.0f / 12.0f) * INV_DX, (-8.0f / 12.0f) * INV_DX, 0.0f,
                   (8.0f / 12.0f) * INV_DX, (-1.0f / 12.0f) * INV_DX);
  }
  return c;
}

__global__ __launch_bounds__(128) void rcnn_step_wmma(
    const float* __restrict__ h,
    const float* __restrict__ p_nu_u, const float* __restrict__ p_nu_v,
    const float* __restrict__ p_c1u, const float* __restrict__ p_c2u,
    const float* __restrict__ p_c1v, const float* __restrict__ p_c2v,
    float* __restrict__ out) {
  const int lane = threadIdx.x;                           // 0..31 (wave32)
  const int ty   = blockIdx.y * blockDim.y + threadIdx.y; // tile row 0..127
  const int tx   = blockIdx.x;                            // tile col 0..127
  const int b    = blockIdx.z;                            // batch 0..7

  const int x0 = tx * 16;
  const int y0 = ty * 16;

  const float nu_u = *p_nu_u, nu_v = *p_nu_v;
  const float c1u = *p_c1u, c2u = *p_c2u;
  const float c1v = *p_c1v, c2v = *p_c2v;

  const size_t slab = (size_t)GN * (size_t)GN;
  const float* __restrict__ su = h + ((size_t)b * 2 + 0) * slab;
  const float* __restrict__ sv = h + ((size_t)b * 2 + 1) * slab;
  float* __restrict__ du = out + ((size_t)b * 2 + 0) * slab;
  float* __restrict__ dv = out + ((size_t)b * 2 + 1) * slab;

  const int n  = lane & 15;  // column within tile (also row m on the A side)
  const int hi = lane >> 4;  // half-wave select

  // Banded coefficient operands: lane-only dependent, shared by the A-side
  // (y-conv) and B-side (x-conv) and by both channels. Computed once, then
  // pinned in VGPRs with an empty asm barrier so LLVM cannot rematerialize
  // the cndmask select chains at every WMMA use site.
  v2f cl[5], cd[5];
#pragma unroll
  for (int j = 0; j < 5; ++j) {
    cl[j] = coef_chunk<true>(j, lane);   // laplacian arm
    cd[j] = coef_chunk<false>(j, lane);  // derivative
    asm volatile("" : "+v"(cl[j]), "+v"(cd[j]));
  }

  // Center values of both channels in C/D layout:
  // VGPR r, lanes 0-15 -> M=r, lanes 16-31 -> M=r+8.
  v8f uc, vc;
#pragma unroll
  for (int r = 0; r < 8; ++r) {
    const int row = y0 + r + (hi << 3);
    uc[r] = su[(size_t)row * GN + (x0 + n)];
    vc[r] = sv[(size_t)row * GN + (x0 + n)];
  }

  v8f lap[2], ddx[2], ddy[2];
#pragma unroll
  for (int ch = 0; ch < 2; ++ch) {
    const float* __restrict__ src = ch ? sv : su;
    v8f L = {};
    v8f Dx = {};
    v8f Dy = {};

    // ---- y-direction convolutions:  D = A_coef(16x20) * X(20x16) ----
    // X(k, n) = h(y0-2+k, x0+n); row index wraps, column never does here.
#pragma unroll
    for (int j = 0; j < 5; ++j) {
      const int col = x0 + n;
      const int rb  = y0 - 2 + 4 * j + (hi << 1);
      v2f xb;
      xb[0] = src[(size_t)((rb) & GMSK) * GN + col];
      xb[1] = src[(size_t)((rb + 1) & GMSK) * GN + col];
      L  = __builtin_amdgcn_wmma_f32_16x16x4_f32(false, cl[j], false, xb,
                                                 (short)0, L, false, false);
      Dx = __builtin_amdgcn_wmma_f32_16x16x4_f32(false, cd[j], false, xb,
                                                 (short)0, Dx, false, false);
    }

    // ---- x-direction convolutions:  D = X(16x20) * B_coef(20x16) ----
    // X(m, k) = h(y0+m, x0-2+k); column index wraps, row never does here.
#pragma unroll
    for (int j = 0; j < 5; ++j) {
      const int row = y0 + n;  // A-operand row m = lane&15
      const int cb  = x0 - 2 + 4 * j + (hi << 1);
      v2f xa;
      xa[0] = src[(size_t)row * GN + ((cb) & GMSK)];
      xa[1] = src[(size_t)row * GN + ((cb + 1) & GMSK)];
      L  = __builtin_amdgcn_wmma_f32_16x16x4_f32(false, xa, false, cl[j],
                                                 (short)0, L, false, false);
      Dy = __builtin_amdgcn_wmma_f32_16x16x4_f32(false, xa, false, cd[j],
                                                 (short)0, Dy, false, false);
    }

    lap[ch] = L;
    ddx[ch] = Dx;
    ddy[ch] = Dy;
  }

  // Elementwise combine in C/D layout + non-temporal store (output is
  // write-once; keep L2 for the input halo reuse).
  // f_u = nu_u*lap_u + C1_u*u*ddx_u + C2_u*v*ddy_u ; out = h + DT*f
#pragma unroll
  for (int r = 0; r < 8; ++r) {
    const int row = y0 + r + (hi << 3);
    const size_t o = (size_t)row * GN + (x0 + n);
    const float u = uc[r];
    const float v = vc[r];
    const float ou = u + DT_C * (nu_u * lap[0][r] + c1u * u * ddx[0][r] +
                                 c2u * v * ddy[0][r]);
    const float ov = v + DT_C * (nu_v * lap[1][r] + c1v * u * ddx[1][r] +
                                 c2v * v * ddy[1][r]);
    __builtin_nontemporal_store(ou, &du[o]);
    __builtin_nontemporal_store(ov, &dv[o]);
  }
}

extern "C" void kernel_launch(void* const* d_in, const int* in_sizes, int n_in,
                              void* d_out, int out_size, void* d_ws,
                              size_t ws_size, hipStream_t stream) {
  (void)in_sizes; (void)n_in; (void)out_size; (void)d_ws; (void)ws_size;
  // setup_inputs order: h, nu_u, nu_v, C1_u, C2_u, C1_v, C2_v
  const float* h = (const float*)d_in[0];
  dim3 block(32, 4, 1);   // 4 wave32 waves, one 16x16 tile each
  dim3 grid(128, 32, 8);  // 128 x (32*4) tiles, 8 batches
  rcnn_step_wmma<<<grid, block, 0, stream>>>(
      h, (const float*)d_in[1], (const float*)d_in[2], (const float*)d_in[3],
      (const float*)d_in[4], (const float*)d_in[5], (const float*)d_in[6],
      (float*)d_out);
}